// FusedLinear_36515811951163
// MI455X (gfx1250) — compile-verified
//
#include <hip/hip_runtime.h>

typedef __attribute__((ext_vector_type(2))) float v2f;
typedef __attribute__((ext_vector_type(8))) float v8f;
typedef __attribute__((ext_vector_type(4))) int   v4i;

#define D_IN   4096
#define D_OUT  4096
#define N_TOK  8192
#define NE     4
#define NR     8
#define ER     32        // NE*NR
#define EPS    1e-6f
#define LORA_SCALE 0.1f

#define KC     64                // K chunk staged in LDS
#define BSTRIDE 68               // padded row stride (dwords): conflict-free ds_load_b64
#define NCOLS  64                // N columns per block

#if __has_builtin(__builtin_amdgcn_global_load_async_to_lds_b128)
#define HAVE_ASYNC_LDS 1
#else
#define HAVE_ASYNC_LDS 0
#endif

typedef __attribute__((address_space(1))) v4i* gv4i_ptr;   // global int32x4*
typedef __attribute__((address_space(3))) v4i* lv4i_ptr;   // LDS    int32x4*

static __device__ __forceinline__ void wait_asynccnt0() {
#if __has_builtin(__builtin_amdgcn_s_wait_asynccnt)
    __builtin_amdgcn_s_wait_asynccnt(0);
#else
    asm volatile("s_wait_asynccnt 0x0" ::: "memory");
#endif
}

static __device__ __forceinline__ v8f wmma_f32(v2f a, v2f b, v8f c) {
    return __builtin_amdgcn_wmma_f32_16x16x4_f32(
        /*neg_a=*/false, a, /*neg_b=*/false, b,
        /*c_mod=*/(short)0, c, /*reuse_a=*/false, /*reuse_b=*/false);
}

// ---------------------------------------------------------------------------
// Kernel 1: qbsfT[o*32 + er] = qb[e][o][r] * sf[e][o]   (er = e*8 + r)
// ---------------------------------------------------------------------------
__global__ __launch_bounds__(256) void build_qbsfT_kernel(
    const float* __restrict__ qb, const float* __restrict__ sf,
    float* __restrict__ qbsfT) {
    int idx = blockIdx.x * blockDim.x + threadIdx.x;   // idx = o*32 + er
    if (idx >= D_OUT * ER) return;
    int o  = idx >> 5;
    int er = idx & 31;
    int e  = er >> 3;
    int r  = er & 7;
    qbsfT[idx] = qb[((size_t)e * D_OUT + o) * NR + r] * sf[(size_t)e * D_OUT + o];
}

// ---------------------------------------------------------------------------
// Kernel 2: inter_s[n][er] = 0.1 * mask(w[n][e]) * sum_d x[n][d]*qa[er][d]
// One wave computes a 32(token) x 32(er) tile via WMMA.
// ---------------------------------------------------------------------------
__global__ __launch_bounds__(256) void lora_a_kernel(
    const float* __restrict__ x, const float* __restrict__ tkw,
    const float* __restrict__ qa, float* __restrict__ inter) {
    int wave = (blockIdx.x * blockDim.x + threadIdx.x) >> 5;  // 0..255
    int lane = threadIdx.x & 31;
    int mBase = wave * 32;
    int half  = lane >> 4;
    int l16   = lane & 15;
    int koff  = half * 2;

    const float* aRow0 = x  + (size_t)(mBase + l16)      * D_IN + koff;
    const float* aRow1 = x  + (size_t)(mBase + 16 + l16) * D_IN + koff;
    const float* bR0   = qa + (size_t)(l16)      * D_IN + koff;   // er = l16
    const float* bR1   = qa + (size_t)(l16 + 16) * D_IN + koff;   // er = l16+16

    v8f acc00 = {}, acc01 = {}, acc10 = {}, acc11 = {};
    for (int k = 0; k < D_IN; k += 4) {
        v2f a0 = *(const v2f*)(aRow0 + k);
        v2f a1 = *(const v2f*)(aRow1 + k);
        v2f b0 = *(const v2f*)(bR0 + k);
        v2f b1 = *(const v2f*)(bR1 + k);
        acc00 = wmma_f32(a0, b0, acc00);
        acc01 = wmma_f32(a0, b1, acc01);
        acc10 = wmma_f32(a1, b0, acc10);
        acc11 = wmma_f32(a1, b1, acc11);
    }

    int er0 = l16, er1 = l16 + 16;
    int e0 = er0 >> 3, e1 = er1 >> 3;
#pragma unroll
    for (int j = 0; j < 8; ++j) {
        int row0 = mBase + j + half * 8;
        int row1 = row0 + 16;
        float w00 = tkw[(size_t)row0 * NE + e0];
        float w01 = tkw[(size_t)row0 * NE + e1];
        float w10 = tkw[(size_t)row1 * NE + e0];
        float w11 = tkw[(size_t)row1 * NE + e1];
        float m00 = (__builtin_fabsf(w00) > EPS) ? w00 : 0.0f;
        float m01 = (__builtin_fabsf(w01) > EPS) ? w01 : 0.0f;
        float m10 = (__builtin_fabsf(w10) > EPS) ? w10 : 0.0f;
        float m11 = (__builtin_fabsf(w11) > EPS) ? w11 : 0.0f;
        inter[(size_t)row0 * ER + er0] = acc00[j] * (LORA_SCALE * m00);
        inter[(size_t)row0 * ER + er1] = acc01[j] * (LORA_SCALE * m01);
        inter[(size_t)row1 * ER + er0] = acc10[j] * (LORA_SCALE * m10);
        inter[(size_t)row1 * ER + er1] = acc11[j] * (LORA_SCALE * m11);
    }
}

// ---------------------------------------------------------------------------
// Kernel 3: out = x @ W^T + inter_s @ qbsfT^T
// Block tile: 256M (8 waves x 32 rows) x 64N. B (W panel) staged through a
// double-buffered LDS tile [64 cols][KC k], padded stride 68 -> ds_load_b64
// operand fetch is bank-conflict free. Copy-in uses the gfx1250 async
// global->LDS engine (ASYNCcnt) when available; one barrier per K-chunk.
// ---------------------------------------------------------------------------
__global__ __launch_bounds__(256) void gemm_kernel(
    const float* __restrict__ x, const float* __restrict__ W,
    const float* __restrict__ inter, const float* __restrict__ qbsfT,
    float* __restrict__ out) {
    __shared__ float bsh[2][NCOLS * BSTRIDE];   // 2 * 17408 B

    int nGroup = blockIdx.x & 63;        // 64 column-groups of width 64
    int mBlock = blockIdx.x >> 6;        // 0..31
    int oBase  = nGroup * NCOLS;

    int wv   = threadIdx.x >> 5;         // 0..7
    int lane = threadIdx.x & 31;
    int mBase = mBlock * 256 + wv * 32;
    int half  = lane >> 4;
    int l16   = lane & 15;
    int koff  = half * 2;

    // --- copy-in mapping: thread t covers col = t/4, k = (t%4)*4 + 16*i ---
    int cCol = threadIdx.x >> 2;
    int kq4  = (threadIdx.x & 3) * 4;
    const float* gsrc = W + (size_t)(oBase + cCol) * D_IN + kq4;
    int ldsOff = cCol * BSTRIDE + kq4;   // dwords

    const float* aRow0 = x + (size_t)(mBase + l16)      * D_IN + koff;
    const float* aRow1 = x + (size_t)(mBase + 16 + l16) * D_IN + koff;

    v8f acc00 = {}, acc01 = {}, acc02 = {}, acc03 = {};
    v8f acc10 = {}, acc11 = {}, acc12 = {}, acc13 = {};

    const int nChunks = D_IN / KC;       // 64

#if HAVE_ASYNC_LDS
    // prologue: issue async copy of chunk 0 into buffer 0
#pragma unroll
    for (int i = 0; i < 4; ++i)
        __builtin_amdgcn_global_load_async_to_lds_b128(
            (gv4i_ptr)(gsrc + 16 * i),
            (lv4i_ptr)(&bsh[0][ldsOff + 16 * i]), 0, 0);

    for (int cb = 0; cb < nChunks; ++cb) {
        int buf = cb & 1;
        wait_asynccnt0();        // own chunk-cb writes landed in LDS
        __syncthreads();         // everyone's writes landed, prev compute done
        if (cb + 1 < nChunks) {  // overlap next copy with this chunk's WMMAs
            const float* g = gsrc + (size_t)(cb + 1) * KC;
#pragma unroll
            for (int i = 0; i < 4; ++i)
                __builtin_amdgcn_global_load_async_to_lds_b128(
                    (gv4i_ptr)(g + 16 * i),
                    (lv4i_ptr)(&bsh[buf ^ 1][ldsOff + 16 * i]), 0, 0);
        }
        const float* bs = &bsh[buf][koff];
        const float* a0p = aRow0 + cb * KC;
        const float* a1p = aRow1 + cb * KC;
#pragma unroll
        for (int kk = 0; kk < KC; kk += 4) {
            v2f a0 = *(const v2f*)(a0p + kk);
            v2f a1 = *(const v2f*)(a1p + kk);
            v2f b0 = *(const v2f*)(bs + (size_t)(l16)      * BSTRIDE + kk);
            v2f b1 = *(const v2f*)(bs + (size_t)(l16 + 16) * BSTRIDE + kk);
            v2f b2 = *(const v2f*)(bs + (size_t)(l16 + 32) * BSTRIDE + kk);
            v2f b3 = *(const v2f*)(bs + (size_t)(l16 + 48) * BSTRIDE + kk);
            acc00 = wmma_f32(a0, b0, acc00);
            acc01 = wmma_f32(a0, b1, acc01);
            acc02 = wmma_f32(a0, b2, acc02);
            acc03 = wmma_f32(a0, b3, acc03);
            acc10 = wmma_f32(a1, b0, acc10);
            acc11 = wmma_f32(a1, b1, acc11);
            acc12 = wmma_f32(a1, b2, acc12);
            acc13 = wmma_f32(a1, b3, acc13);
        }
    }
#else
    // fallback: register-buffered copy (load -> ds_store), one barrier/chunk
    float* lds_dst = &bsh[0][ldsOff];
    float4 creg[4];
#pragma unroll
    for (int i = 0; i < 4; ++i)
        creg[i] = *(const float4*)(gsrc + 16 * i);

    for (int cb = 0; cb < nChunks; ++cb) {
        int buf = cb & 1;
        float* dst = lds_dst + buf * (NCOLS * BSTRIDE);
#pragma unroll
        for (int i = 0; i < 4; ++i)
            *(float4*)(dst + 16 * i) = creg[i];
        if (cb + 1 < nChunks) {
            const float* g = gsrc + (size_t)(cb + 1) * KC;
#pragma unroll
            for (int i = 0; i < 4; ++i)
                creg[i] = *(const float4*)(g + 16 * i);
        }
        __syncthreads();

        const float* bs = &bsh[buf][koff];
        const float* a0p = aRow0 + cb * KC;
        const float* a1p = aRow1 + cb * KC;
#pragma unroll
        for (int kk = 0; kk < KC; kk += 4) {
            v2f a0 = *(const v2f*)(a0p + kk);
            v2f a1 = *(const v2f*)(a1p + kk);
            v2f b0 = *(const v2f*)(bs + (size_t)(l16)      * BSTRIDE + kk);
            v2f b1 = *(const v2f*)(bs + (size_t)(l16 + 16) * BSTRIDE + kk);
            v2f b2 = *(const v2f*)(bs + (size_t)(l16 + 32) * BSTRIDE + kk);
            v2f b3 = *(const v2f*)(bs + (size_t)(l16 + 48) * BSTRIDE + kk);
            acc00 = wmma_f32(a0, b0, acc00);
            acc01 = wmma_f32(a0, b1, acc01);
            acc02 = wmma_f32(a0, b2, acc02);
            acc03 = wmma_f32(a0, b3, acc03);
            acc10 = wmma_f32(a1, b0, acc10);
            acc11 = wmma_f32(a1, b1, acc11);
            acc12 = wmma_f32(a1, b2, acc12);
            acc13 = wmma_f32(a1, b3, acc13);
        }
    }
#endif

    // Adapter as a K-extension of the same GEMM (K = 32), operands from global
    const float* aRow20 = inter + (size_t)(mBase + l16)      * ER + koff;
    const float* aRow21 = inter + (size_t)(mBase + 16 + l16) * ER + koff;
    const float* bCol2  = qbsfT + (size_t)(oBase + l16)      * ER + koff;
#pragma unroll
    for (int k = 0; k < ER; k += 4) {
        v2f a0 = *(const v2f*)(aRow20 + k);
        v2f a1 = *(const v2f*)(aRow21 + k);
        v2f b0 = *(const v2f*)(bCol2 + k);
        v2f b1 = *(const v2f*)(bCol2 + 16 * ER + k);
        v2f b2 = *(const v2f*)(bCol2 + 32 * ER + k);
        v2f b3 = *(const v2f*)(bCol2 + 48 * ER + k);
        acc00 = wmma_f32(a0, b0, acc00);
        acc01 = wmma_f32(a0, b1, acc01);
        acc02 = wmma_f32(a0, b2, acc02);
        acc03 = wmma_f32(a0, b3, acc03);
        acc10 = wmma_f32(a1, b0, acc10);
        acc11 = wmma_f32(a1, b1, acc11);
        acc12 = wmma_f32(a1, b2, acc12);
        acc13 = wmma_f32(a1, b3, acc13);
    }

#pragma unroll
    for (int j = 0; j < 8; ++j) {
        int row0 = mBase + j + half * 8;
        int row1 = row0 + 16;
        float* orow0 = out + (size_t)row0 * D_OUT + oBase + l16;
        float* orow1 = out + (size_t)row1 * D_OUT + oBase + l16;
        orow0[0]  = acc00[j];
        orow0[16] = acc01[j];
        orow0[32] = acc02[j];
        orow0[48] = acc03[j];
        orow1[0]  = acc10[j];
        orow1[16] = acc11[j];
        orow1[32] = acc12[j];
        orow1[48] = acc13[j];
    }
}

// ---------------------------------------------------------------------------
extern "C" void kernel_launch(void* const* d_in, const int* in_sizes, int n_in,
                              void* d_out, int out_size, void* d_ws, size_t ws_size,
                              hipStream_t stream) {
    const float* x   = (const float*)d_in[0];
    const float* tkw = (const float*)d_in[1];
    const float* W   = (const float*)d_in[2];
    const float* qa  = (const float*)d_in[3];
    const float* qb  = (const float*)d_in[4];
    const float* sf  = (const float*)d_in[5];
    float* out = (float*)d_out;

    float* inter = (float*)d_ws;                                   // 8192*32*4 = 1 MB
    float* qbsfT = (float*)((char*)d_ws + (size_t)N_TOK * ER * 4); // 4096*32*4 = 512 KB

    build_qbsfT_kernel<<<(D_OUT * ER + 255) / 256, 256, 0, stream>>>(qb, sf, qbsfT);

    // 256 waves (32 tokens each), 8 waves per block
    lora_a_kernel<<<(N_TOK / 32) / 8, 256, 0, stream>>>(x, tkw, qa, inter);

    // (8192/256) M-blocks * (4096/64) N-groups = 2048 blocks, 256 threads
    gemm_kernel<<<(N_TOK / 256) * (D_OUT / NCOLS), 256, 0, stream>>>(x, W, inter, qbsfT, out);
}